// SpiralAutoencoder_55327768707353
// MI455X (gfx1250) — compile-verified
//
#include <hip/hip_runtime.h>
#include <hip/hip_bf16.h>

// ---------------------------------------------------------------------------
// SpiralNet autoencoder forward on gfx1250 (CDNA5), fp32, all GEMM-shaped ops
// via V_WMMA_F32_16X16X4_F32 (exact fp32 semantics). wave32.
// Each wave holds up to 8 independent 16x16 accumulators and is software-
// pipelined: chunk k+1 fragment loads are issued before chunk k's WMMA chain,
// so global loads overlap the matrix pipe and per-chunk waits shrink.
//
// Fragment layouts (ISA 7.12.2):
//   A (16x4):  lane l -> row l&15, k = kbase + (l>>4)*2 + j   (j = VGPR 0/1)
//   B (4x16):  lane l -> col l&15, k = kbase + (l>>4)*2 + j
//   C/D(16x16):VGPR v, lane l -> row v + 8*(l>>4), col l&15
// ---------------------------------------------------------------------------

typedef float v2f __attribute__((ext_vector_type(2)));
typedef float v8f __attribute__((ext_vector_type(8)));

#define SPIRAL_K 12

__device__ __forceinline__ float elu_act(float v) {
    return v > 0.0f ? v : (__expf(v) - 1.0f);
}

__device__ __forceinline__ v8f wmma_f32(v2f a, v2f b, v8f c) {
    return __builtin_amdgcn_wmma_f32_16x16x4_f32(false, a, false, b,
                                                 (short)0, c, false, false);
}

// ---------------------------------------------------------------------------
// Spiral conv: y[b,n,o] = act( sum_{k,f} x[b, S[n,k], f] * W[o, k*FIN+f] + b[o] )
// row N-1 (padding vertex) zeroed. One wave: 16 rows x (16*OT) out cols.
// KF = 12*FIN is a multiple of 4 -> no K tail. Pipelined gather.
// ---------------------------------------------------------------------------
template <int ACT, int FIN, int OT>
__global__ __launch_bounds__(32)
void spiral_conv_wmma(const float* __restrict__ x, const int* __restrict__ S,
                      const float* __restrict__ W, const float* __restrict__ bias,
                      float* __restrict__ y, int N, int Fout)
{
    constexpr int KF = SPIRAL_K * FIN;
    const int lane = threadIdx.x;
    const int half = lane >> 4;
    const int lr   = lane & 15;

    const int n0 = blockIdx.x * 16;
    const int o0 = blockIdx.y * (16 * OT);
    const int b  = blockIdx.z;

    int row = n0 + lr;
    if (row > N - 1) row = N - 1;
    int sidx[SPIRAL_K];
#pragma unroll
    for (int k = 0; k < SPIRAL_K; ++k) sidx[k] = S[row * SPIRAL_K + k];

    const float* xb = x + (size_t)b * N * FIN;

    bool colok[OT];
    const float* wr[OT];
#pragma unroll
    for (int t = 0; t < OT; ++t) {
        const int o = o0 + t * 16 + lr;
        colok[t] = o < Fout;
        wr[t] = W + (size_t)(colok[t] ? o : 0) * KF + half * 2;
    }

    v8f c[OT];
#pragma unroll
    for (int t = 0; t < OT; ++t) c[t] = (v8f){0.f,0.f,0.f,0.f,0.f,0.f,0.f,0.f};

    // gather one A fragment pair for k-chunk kb
    auto gatherA = [&](int kb) -> v2f {
        const int kk = kb + half * 2;
        v2f a;
        if constexpr ((FIN & 1) == 0) {
            a = *(const v2f*)(xb + (size_t)sidx[kk / FIN] * FIN + (kk % FIN));
        } else {
            a.x = xb[(size_t)sidx[kk / FIN] * FIN + (kk % FIN)];
            a.y = xb[(size_t)sidx[(kk + 1) / FIN] * FIN + ((kk + 1) % FIN)];
        }
        return a;
    };

    // ---- software pipeline: preload chunk 0 ----
    v2f a = gatherA(0);
    v2f bw[OT];
#pragma unroll
    for (int t = 0; t < OT; ++t) {
        v2f bb = *(const v2f*)(wr[t]);
        bb.x = colok[t] ? bb.x : 0.f;
        bb.y = colok[t] ? bb.y : 0.f;
        bw[t] = bb;
    }

    for (int kb = 0; kb + 4 < KF; kb += 4) {
        v2f an = gatherA(kb + 4);
        v2f bn[OT];
#pragma unroll
        for (int t = 0; t < OT; ++t) {
            v2f bb = *(const v2f*)(wr[t] + kb + 4);
            bb.x = colok[t] ? bb.x : 0.f;
            bb.y = colok[t] ? bb.y : 0.f;
            bn[t] = bb;
        }
#pragma unroll
        for (int t = 0; t < OT; ++t) c[t] = wmma_f32(a, bw[t], c[t]);
        a = an;
#pragma unroll
        for (int t = 0; t < OT; ++t) bw[t] = bn[t];
    }
#pragma unroll
    for (int t = 0; t < OT; ++t) c[t] = wmma_f32(a, bw[t], c[t]);

    // ---- epilogue ----
#pragma unroll
    for (int t = 0; t < OT; ++t) {
        const int o  = o0 + t * 16 + lr;
        const float bv = colok[t] ? bias[o] : 0.f;
#pragma unroll
        for (int v = 0; v < 8; ++v) {
            const int gn = n0 + v + half * 8;
            if (gn < N && colok[t]) {
                float val = c[t][v] + bv;
                if (ACT) val = elu_act(val);
                if (gn == N - 1) val = 0.f;       // padding-vertex mask
                y[(size_t)b * N * Fout + (size_t)gn * Fout + o] = val;
            }
        }
    }
}

// ---------------------------------------------------------------------------
// Down/up-sample: y[b,m,f] = sum_n M[m,n] * x[b,n,f]
// One wave: (16*MT) m-rows x (16*FT) f-cols. MT*FT accumulators; A fragments
// reused across FT f-tiles, B fragments across MT m-tiles. Pipelined.
// F is a template constant -> immediate strides.
// ---------------------------------------------------------------------------
template <int MT, int FT, int F>
__global__ __launch_bounds__(32)
void dense_map_wmma(const float* __restrict__ M, const float* __restrict__ x,
                    float* __restrict__ y, int Mo, int Ni)
{
    const int lane = threadIdx.x;
    const int half = lane >> 4;
    const int lr   = lane & 15;

    const int m0 = blockIdx.x * (16 * MT);
    const int f0 = blockIdx.y * (16 * FT);
    const int b  = blockIdx.z;

    const float* mrow[MT];
    const float* mp[MT];
#pragma unroll
    for (int t = 0; t < MT; ++t) {
        int m = m0 + t * 16 + lr;
        if (m > Mo - 1) m = Mo - 1;
        mrow[t] = M + (size_t)m * Ni;
        mp[t]   = mrow[t] + half * 2;
    }
    const float* xrow = x + (size_t)b * Ni * F + f0 + lr;
    const float* xp   = xrow + (half * 2) * F;

    v8f c[MT][FT];
#pragma unroll
    for (int t = 0; t < MT; ++t)
#pragma unroll
        for (int ft = 0; ft < FT; ++ft)
            c[t][ft] = (v8f){0.f,0.f,0.f,0.f,0.f,0.f,0.f,0.f};

    const int Ni4 = Ni & ~3;

    // ---- software pipeline: preload chunk 0 ----
    v2f a[MT], bb[FT];
#pragma unroll
    for (int t = 0; t < MT; ++t) a[t] = *(const v2f*)mp[t];
#pragma unroll
    for (int ft = 0; ft < FT; ++ft) {
        bb[ft].x = xp[ft * 16];
        bb[ft].y = xp[F + ft * 16];
    }

    for (int kb = 0; kb + 4 < Ni4; kb += 4) {
        v2f an[MT], bn[FT];
#pragma unroll
        for (int t = 0; t < MT; ++t) an[t] = *(const v2f*)(mp[t] + 4);
#pragma unroll
        for (int ft = 0; ft < FT; ++ft) {
            bn[ft].x = xp[4 * F + ft * 16];
            bn[ft].y = xp[5 * F + ft * 16];
        }
#pragma unroll
        for (int t = 0; t < MT; ++t)
#pragma unroll
            for (int ft = 0; ft < FT; ++ft)
                c[t][ft] = wmma_f32(a[t], bb[ft], c[t][ft]);
#pragma unroll
        for (int t = 0; t < MT; ++t) { a[t] = an[t]; mp[t] += 4; }
#pragma unroll
        for (int ft = 0; ft < FT; ++ft) bb[ft] = bn[ft];
        xp += 4 * F;
    }
#pragma unroll
    for (int t = 0; t < MT; ++t)
#pragma unroll
        for (int ft = 0; ft < FT; ++ft)
            c[t][ft] = wmma_f32(a[t], bb[ft], c[t][ft]);

    if (Ni4 < Ni) {                  // uniform K-tail: clamped loads + selects
        const int kk = Ni4 + half * 2;
        const int k0 = kk     < Ni ? kk     : Ni - 1;
        const int k1 = kk + 1 < Ni ? kk + 1 : Ni - 1;
        v2f bt[FT];
#pragma unroll
        for (int ft = 0; ft < FT; ++ft) {
            bt[ft].x = (kk     < Ni) ? xrow[(size_t)k0 * F + ft * 16] : 0.f;
            bt[ft].y = (kk + 1 < Ni) ? xrow[(size_t)k1 * F + ft * 16] : 0.f;
        }
#pragma unroll
        for (int t = 0; t < MT; ++t) {
            v2f at;
            at.x = (kk     < Ni) ? mrow[t][k0] : 0.f;
            at.y = (kk + 1 < Ni) ? mrow[t][k1] : 0.f;
#pragma unroll
            for (int ft = 0; ft < FT; ++ft)
                c[t][ft] = wmma_f32(at, bt[ft], c[t][ft]);
        }
    }

#pragma unroll
    for (int t = 0; t < MT; ++t)
#pragma unroll
        for (int ft = 0; ft < FT; ++ft)
#pragma unroll
            for (int v = 0; v < 8; ++v) {
                const int m = m0 + t * 16 + v + half * 8;
                if (m < Mo)
                    y[(size_t)b * Mo * F + (size_t)m * F + f0 + ft * 16 + lr] = c[t][ft][v];
            }
}

// ---------------------------------------------------------------------------
// Fully-connected: y[r,o] = sum_j X[r,j] * W[o,j] + bias[o]
// JMOD != 0 -> concat([z,z]) input: X[r,j] = Xbase[r*JMOD + (j & (JMOD-1))]
// R multiple of 16; O multiple of 16*OT; J multiple of 4. Pipelined.
// ---------------------------------------------------------------------------
template <int JMOD, int OT>
__global__ __launch_bounds__(32)
void linear_wmma(const float* __restrict__ X, const float* __restrict__ W,
                 const float* __restrict__ bias, float* __restrict__ y,
                 int J, int O)
{
    const int lane = threadIdx.x;
    const int half = lane >> 4;
    const int lr   = lane & 15;

    const int r0 = blockIdx.x * 16;
    const int o0 = blockIdx.y * (16 * OT);

    const int rowStride = JMOD ? JMOD : J;
    const float* xr = X + (size_t)(r0 + lr) * rowStride;
    const float* wr[OT];
#pragma unroll
    for (int t = 0; t < OT; ++t)
        wr[t] = W + (size_t)(o0 + t * 16 + lr) * J + half * 2;

    v8f c[OT];
#pragma unroll
    for (int t = 0; t < OT; ++t) c[t] = (v8f){0.f,0.f,0.f,0.f,0.f,0.f,0.f,0.f};

    auto loadA = [&](int kb) -> v2f {
        const int kk = kb + half * 2;
        const int j0 = JMOD ? (kk & (JMOD - 1)) : kk;   // kk even -> pair in-block
        return *(const v2f*)(xr + j0);
    };

    v2f a = loadA(0);
    v2f bw[OT];
#pragma unroll
    for (int t = 0; t < OT; ++t) bw[t] = *(const v2f*)(wr[t]);

    for (int kb = 0; kb + 4 < J; kb += 4) {
        v2f an = loadA(kb + 4);
        v2f bn[OT];
#pragma unroll
        for (int t = 0; t < OT; ++t) bn[t] = *(const v2f*)(wr[t] + kb + 4);
#pragma unroll
        for (int t = 0; t < OT; ++t) c[t] = wmma_f32(a, bw[t], c[t]);
        a = an;
#pragma unroll
        for (int t = 0; t < OT; ++t) bw[t] = bn[t];
    }
#pragma unroll
    for (int t = 0; t < OT; ++t) c[t] = wmma_f32(a, bw[t], c[t]);

#pragma unroll
    for (int t = 0; t < OT; ++t) {
        const int o = o0 + t * 16 + lr;
        const float bv = bias[o];
#pragma unroll
        for (int v = 0; v < 8; ++v) {
            const int r = r0 + v + half * 8;
            y[(size_t)r * O + o] = c[t][v] + bv;
        }
    }
}

// out2 = out1 + template
__global__ __launch_bounds__(256)
void add_template_kernel(const float* __restrict__ a, const float* __restrict__ t,
                         float* __restrict__ o, int n)
{
    int i = blockIdx.x * blockDim.x + threadIdx.x;
    if (i < n) o[i] = a[i] + t[i];
}

// ---------------------------------------------------------------------------
// Host-side pipeline
// ---------------------------------------------------------------------------
extern "C" void kernel_launch(void* const* d_in, const int* in_sizes, int n_in,
                              void* d_out, int out_size, void* d_ws, size_t ws_size,
                              hipStream_t stream) {
    (void)in_sizes; (void)n_in; (void)out_size; (void)ws_size;

    const int B = 64;
    const int NV0 = 5024, NV1 = 1257, NV2 = 315, NV3 = 80, NV4 = 21;

    const float* x_talking = (const float*)d_in[0];
    const float* templ     = (const float*)d_in[1];
    const float* D0 = (const float*)d_in[2];
    const float* U0 = (const float*)d_in[3];
    const float* D1 = (const float*)d_in[4];
    const float* U1 = (const float*)d_in[5];
    const float* D2 = (const float*)d_in[6];
    const float* U2 = (const float*)d_in[7];
    const float* D3 = (const float*)d_in[8];
    const float* U3 = (const float*)d_in[9];
    const float* enc_w0 = (const float*)d_in[10]; const float* enc_b0 = (const float*)d_in[11];
    const float* enc_w1 = (const float*)d_in[12]; const float* enc_b1 = (const float*)d_in[13];
    const float* enc_w2 = (const float*)d_in[14]; const float* enc_b2 = (const float*)d_in[15];
    const float* enc_w3 = (const float*)d_in[16]; const float* enc_b3 = (const float*)d_in[17];
    const float* fc_enc_w = (const float*)d_in[18]; const float* fc_enc_b = (const float*)d_in[19];
    const float* fc_dec_w = (const float*)d_in[20]; const float* fc_dec_b = (const float*)d_in[21];
    const float* dec_w0 = (const float*)d_in[22]; const float* dec_b0 = (const float*)d_in[23];
    const float* dec_w1 = (const float*)d_in[24]; const float* dec_b1 = (const float*)d_in[25];
    const float* dec_w2 = (const float*)d_in[26]; const float* dec_b2 = (const float*)d_in[27];
    const float* dec_w3 = (const float*)d_in[28]; const float* dec_b3 = (const float*)d_in[29];
    const float* dec_w4 = (const float*)d_in[30]; const float* dec_b4 = (const float*)d_in[31];
    const int* S0 = (const int*)d_in[32];
    const int* S1 = (const int*)d_in[33];
    const int* S2 = (const int*)d_in[34];
    const int* S3 = (const int*)d_in[35];

    float* out1 = (float*)d_out;
    float* out2 = out1 + (size_t)B * NV0 * 3;

    const size_t bufElems = (size_t)B * NV0 * 32;
    float* bufA = (float*)d_ws;
    float* bufB = bufA + bufElems;
    float* zbuf = bufB + bufElems;

    const dim3 wave(32, 1, 1);
    #define TILES(n) (((n) + 15) / 16)
    #define TILES64(n) (((n) + 63) / 64)

    // ---------------- encoder ----------------
    spiral_conv_wmma<1, 3, 1><<<dim3(TILES(NV0), 1, B), wave, 0, stream>>>(
        x_talking, S0, enc_w0, enc_b0, bufA, NV0, 16);                  // [64,5024,16]
    dense_map_wmma<4, 1, 16><<<dim3(TILES64(NV1), 1, B), wave, 0, stream>>>(
        D0, bufA, bufB, NV1, NV0);                                      // [64,1257,16]
    spiral_conv_wmma<1, 16, 2><<<dim3(TILES(NV1), 1, B), wave, 0, stream>>>(
        bufB, S1, enc_w1, enc_b1, bufA, NV1, 32);                       // [64,1257,32]
    dense_map_wmma<4, 2, 32><<<dim3(TILES64(NV2), 1, B), wave, 0, stream>>>(
        D1, bufA, bufB, NV2, NV1);                                      // [64,315,32]
    spiral_conv_wmma<1, 32, 4><<<dim3(TILES(NV2), 1, B), wave, 0, stream>>>(
        bufB, S2, enc_w2, enc_b2, bufA, NV2, 64);                       // [64,315,64]
    dense_map_wmma<4, 2, 64><<<dim3(TILES64(NV3), 2, B), wave, 0, stream>>>(
        D2, bufA, bufB, NV3, NV2);                                      // [64,80,64]
    spiral_conv_wmma<1, 64, 4><<<dim3(TILES(NV3), 2, B), wave, 0, stream>>>(
        bufB, S3, enc_w3, enc_b3, bufA, NV3, 128);                      // [64,80,128]
    dense_map_wmma<4, 2, 128><<<dim3(TILES64(NV4), 4, B), wave, 0, stream>>>(
        D3, bufA, bufB, NV4, NV3);                                      // [64,21,128]

    // ---------------- latent FCs ----------------
    linear_wmma<0, 4><<<dim3(B / 16, 128 / 64), wave, 0, stream>>>(
        bufB, fc_enc_w, fc_enc_b, zbuf, NV4 * 128, 128);                // z [64,128]
    linear_wmma<128, 4><<<dim3(B / 16, (NV4 * 128) / 64), wave, 0, stream>>>(
        zbuf, fc_dec_w, fc_dec_b, bufA, 2 * 128, NV4 * 128);            // h [64,2688]

    // ---------------- decoder ----------------
    dense_map_wmma<4, 2, 128><<<dim3(TILES64(NV3), 4, B), wave, 0, stream>>>(
        U3, bufA, bufB, NV3, NV4);                                      // [64,80,128]
    spiral_conv_wmma<1, 128, 4><<<dim3(TILES(NV3), 1, B), wave, 0, stream>>>(
        bufB, S3, dec_w0, dec_b0, bufA, NV3, 64);                       // [64,80,64]
    dense_map_wmma<4, 2, 64><<<dim3(TILES64(NV2), 2, B), wave, 0, stream>>>(
        U2, bufA, bufB, NV2, NV3);                                      // [64,315,64]
    spiral_conv_wmma<1, 64, 2><<<dim3(TILES(NV2), 1, B), wave, 0, stream>>>(
        bufB, S2, dec_w1, dec_b1, bufA, NV2, 32);                       // [64,315,32]
    dense_map_wmma<4, 2, 32><<<dim3(TILES64(NV1), 1, B), wave, 0, stream>>>(
        U1, bufA, bufB, NV1, NV2);                                      // [64,1257,32]
    spiral_conv_wmma<1, 32, 2><<<dim3(TILES(NV1), 1, B), wave, 0, stream>>>(
        bufB, S1, dec_w2, dec_b2, bufA, NV1, 32);                       // [64,1257,32]
    dense_map_wmma<4, 2, 32><<<dim3(TILES64(NV0), 1, B), wave, 0, stream>>>(
        U0, bufA, bufB, NV0, NV1);                                      // [64,5024,32]
    spiral_conv_wmma<1, 32, 1><<<dim3(TILES(NV0), 1, B), wave, 0, stream>>>(
        bufB, S0, dec_w3, dec_b3, bufA, NV0, 16);                       // [64,5024,16]
    spiral_conv_wmma<0, 16, 1><<<dim3(TILES(NV0), 1, B), wave, 0, stream>>>(
        bufA, S0, dec_w4, dec_b4, out1, NV0, 3);                        // [64,5024,3]

    const int n_out = B * NV0 * 3;
    add_template_kernel<<<(n_out + 255) / 256, 256, 0, stream>>>(
        out1, templ, out2, n_out);

    #undef TILES
    #undef TILES64
}